// AttentionModule_76751065579714
// MI455X (gfx1250) — compile-verified
//
#include <hip/hip_runtime.h>

#define DIM   1024
#define BATCH 128
#define EPS   1e-5f

typedef __attribute__((ext_vector_type(2))) float v2f;
typedef __attribute__((ext_vector_type(8))) float v8f;

// ---------------------------------------------------------------------------
// Kernel 1: q/k/v = z @ W.T via V_WMMA_F32_16X16X4_F32 (full fp32 precision).
// One wave = one 16x16 output tile. blockDim=128 (4 waves, wave32).
// A (16x4 f32): lane L holds A[L&15][2*(L>>4)+{0,1}]  -> float2 load from z
// B (4x16 f32): lane L holds B[2*(L>>4)+{0,1}][L&15]  -> B[k][n]=W[n][k], float2 load from W row
// C/D (16x16 f32, 8 VGPRs): vgpr r, lane L -> D[r + 8*(L>>4)][L&15]
// ---------------------------------------------------------------------------
__global__ __launch_bounds__(128) void gemm_qkv_wmma(
    const float* __restrict__ z,
    const float* __restrict__ Wq,
    const float* __restrict__ Wk,
    const float* __restrict__ Wv,
    float* __restrict__ qws,
    float* __restrict__ ksws,
    float* __restrict__ vws)
{
    const int wave = threadIdx.x >> 5;
    const int lane = threadIdx.x & 31;
    const int mat  = blockIdx.z;                    // 0=q, 1=k, 2=v

    const float* W    = (mat == 0) ? Wq : (mat == 1 ? Wk : Wv);
    float*       outp = (mat == 0) ? qws : (mat == 1 ? ksws : vws);
    const float  scale = (mat == 1) ? (1.0f / 32.0f) : 1.0f;   // k / sqrt(1024)

    const int m0 = blockIdx.y * 16;                          // batch-row tile
    const int n0 = (blockIdx.x * 4 + wave) * 16;             // out-feature tile

    const int mn = lane & 15;                                // m for A, n for B
    const int kb = (lane >> 4) << 1;                         // K sub-offset: 0 or 2

    const float* Ap = z + (size_t)(m0 + mn) * DIM + kb;      // z[m][k..k+1]
    const float* Bp = W + (size_t)(n0 + mn) * DIM + kb;      // W[n][k..k+1] == B[k][n]

    v8f acc = {};
    #pragma unroll 8
    for (int k = 0; k < DIM; k += 4) {
        v2f a = *reinterpret_cast<const v2f*>(Ap + k);
        v2f b = *reinterpret_cast<const v2f*>(Bp + k);
        // (neg_a, A, neg_b, B, c_mod, C, reuse_a, reuse_b)
        acc = __builtin_amdgcn_wmma_f32_16x16x4_f32(false, a, false, b,
                                                    (short)0, acc, false, false);
    }

    const int rowHi = (lane >> 4) * 8;
    #pragma unroll
    for (int r = 0; r < 8; ++r) {
        const int row = m0 + rowHi + r;
        const int col = n0 + mn;
        outp[(size_t)row * DIM + col] = acc[r] * scale;
    }
}

// ---------------------------------------------------------------------------
// Kernel 2: fused outer-product softmax-attention row.
// One block per batch b (128 blocks, 256 threads). ks/v rows staged in LDS.
// out[b,i] = sum_j exp(q_i*ks_j - m_i)*v_j / sum_j exp(q_i*ks_j - m_i) + v_i
// m_i = q_i * (q_i >= 0 ? max_j ks_j : min_j ks_j)   (exact row max -> stable)
// 134M v_exp_f32 total: VALU/transcendental bound, 4 i-rows per lane to
// amortize each LDS read across 4 exp+FMA chains.
// ---------------------------------------------------------------------------
__global__ __launch_bounds__(256) void attention_row_kernel(
    const float* __restrict__ qws,
    const float* __restrict__ ksws,
    const float* __restrict__ vws,
    float* __restrict__ outws)
{
    __shared__ float sk[DIM];
    __shared__ float sv[DIM];
    __shared__ float rmax[256];
    __shared__ float rmin[256];

    const int b   = blockIdx.x;
    const int tid = threadIdx.x;

    float lmax = -3.402823466e38f, lmin = 3.402823466e38f;
    #pragma unroll
    for (int r = 0; r < 4; ++r) {
        const int j = tid + r * 256;
        const float kj = ksws[(size_t)b * DIM + j];
        const float vj = vws[(size_t)b * DIM + j];
        sk[j] = kj;
        sv[j] = vj;
        lmax = fmaxf(lmax, kj);
        lmin = fminf(lmin, kj);
    }
    rmax[tid] = lmax;
    rmin[tid] = lmin;
    __syncthreads();
    for (int s = 128; s > 0; s >>= 1) {
        if (tid < s) {
            rmax[tid] = fmaxf(rmax[tid], rmax[tid + s]);
            rmin[tid] = fminf(rmin[tid], rmin[tid + s]);
        }
        __syncthreads();
    }
    const float kmax = rmax[0];
    const float kmin = rmin[0];

    const float LOG2E = 1.4426950408889634f;
    float qL[4], mL[4], den[4], num[4];
    #pragma unroll
    for (int u = 0; u < 4; ++u) {
        const int i = tid + u * 256;
        const float qi = qws[(size_t)b * DIM + i];
        qL[u]  = qi * LOG2E;                          // exp(x) = exp2(x*log2e)
        mL[u]  = qL[u] * (qi >= 0.0f ? kmax : kmin);  // = m_i * log2e
        den[u] = 0.0f;
        num[u] = 0.0f;
    }

    #pragma unroll 4
    for (int j = 0; j < DIM; ++j) {
        const float kj = sk[j];
        const float vj = sv[j];
        #pragma unroll
        for (int u = 0; u < 4; ++u) {
            const float e = __builtin_amdgcn_exp2f(qL[u] * kj - mL[u]); // v_exp_f32
            den[u] += e;
            num[u] = fmaf(e, vj, num[u]);
        }
    }

    #pragma unroll
    for (int u = 0; u < 4; ++u) {
        const int i = tid + u * 256;
        outws[(size_t)b * DIM + i] = num[u] / den[u] + sv[i];
    }
}

// ---------------------------------------------------------------------------
// Kernel 3: BatchNorm1d (training-mode batch stats, biased variance).
// One thread per feature; two passes over the 128-row column (L2 resident).
// ---------------------------------------------------------------------------
__global__ __launch_bounds__(256) void batchnorm_kernel(
    const float* __restrict__ outws,
    const float* __restrict__ gamma,
    const float* __restrict__ beta,
    float* __restrict__ out)
{
    const int f = blockIdx.x * 256 + threadIdx.x;
    float sum = 0.0f, sq = 0.0f;
    #pragma unroll 4
    for (int bb = 0; bb < BATCH; ++bb) {
        const float x = outws[(size_t)bb * DIM + f];
        sum += x;
        sq = fmaf(x, x, sq);
    }
    const float mean = sum * (1.0f / BATCH);
    const float var  = sq * (1.0f / BATCH) - mean * mean;
    const float g    = gamma[f] * rsqrtf(var + EPS);
    const float bt   = beta[f];
    #pragma unroll 4
    for (int bb = 0; bb < BATCH; ++bb) {
        const float x = outws[(size_t)bb * DIM + f];
        out[(size_t)bb * DIM + f] = (x - mean) * g + bt;
    }
}

// ---------------------------------------------------------------------------
extern "C" void kernel_launch(void* const* d_in, const int* in_sizes, int n_in,
                              void* d_out, int out_size, void* d_ws, size_t ws_size,
                              hipStream_t stream) {
    (void)in_sizes; (void)n_in; (void)out_size; (void)ws_size;

    const float* z     = (const float*)d_in[0];
    const float* Wq    = (const float*)d_in[1];
    const float* Wk    = (const float*)d_in[2];
    const float* Wv    = (const float*)d_in[3];
    const float* gamma = (const float*)d_in[4];
    const float* beta  = (const float*)d_in[5];
    float* out = (float*)d_out;

    float* ws    = (float*)d_ws;               // 4 * 128*1024 f32 = 2 MB scratch
    float* qws   = ws;
    float* ksws  = ws + (size_t)BATCH * DIM;
    float* vws   = ws + (size_t)2 * BATCH * DIM;
    float* outws = ws + (size_t)3 * BATCH * DIM;

    // x: 16 blocks * 4 waves * 16 = 1024 features; y: 128/16 = 8 batch tiles; z: q/k/v
    dim3 g1(16, 8, 3);
    gemm_qkv_wmma<<<g1, 128, 0, stream>>>(z, Wq, Wk, Wv, qws, ksws, vws);

    attention_row_kernel<<<BATCH, 256, 0, stream>>>(qws, ksws, vws, outws);

    batchnorm_kernel<<<DIM / 256, 256, 0, stream>>>(outws, gamma, beta, out);
}